// AssetGCN_5488968204379
// MI455X (gfx1250) — compile-verified
//
#include <hip/hip_runtime.h>
#include <hip/hip_bf16.h>

// ---------------------------------------------------------------------------
// AssetGCN on MI455X (gfx1250): fp32 WMMA (v_wmma_f32_16x16x4_f32) for both
// GEMM stages of each GCN layer; fused 1x3 conv pair (avoids 838MB temp).
// Round 3: 32x32 output tile per wave (4 acc chains, B fragments shared by
// both m-halves -> 1.5 loads/WMMA), uniform-branch store epilogues with
// immediate-offset stores, row-clamped (branch-free) fragment loads.
// ---------------------------------------------------------------------------

#define G_   128
#define A_   100
#define WIN_ 10
#define FD_  16
#define H_   128
#define F1_  (WIN_ * FD_)   // 160
#define N_   (G_ * A_)      // 12800

typedef __attribute__((ext_vector_type(2))) float v2f;
typedef __attribute__((ext_vector_type(8))) float v8f;

// ---------------------------------------------------------------------------
// Kernel 1: per-group correlation -> normalized adjacency  [G,100,100]
// ---------------------------------------------------------------------------
__global__ __launch_bounds__(128) void corr_kernel(const float* __restrict__ x,
                                                   float* __restrict__ nadj) {
    __shared__ float xc[A_][WIN_];   // centered returns
    __shared__ float dinv[A_];       // 1/sqrt(diag cov)
    __shared__ float rs[A_];         // 1/sqrt(adj row sum)

    const int g   = blockIdx.x;
    const int tid = threadIdx.x;
    const float* xg = x + (size_t)g * A_ * F1_;

    for (int a = tid; a < A_; a += 128) {
        float r[WIN_]; float mean = 0.0f;
        #pragma unroll
        for (int t = 0; t < WIN_; ++t) {
            r[t] = xg[a * F1_ + t * FD_ + (FD_ - 1)];
            mean += r[t];
        }
        mean *= (1.0f / WIN_);
        float ss = 0.0f;
        #pragma unroll
        for (int t = 0; t < WIN_; ++t) {
            float v = r[t] - mean;
            xc[a][t] = v;
            ss += v * v;
        }
        dinv[a] = rsqrtf(ss);
    }
    __syncthreads();

    for (int i = tid; i < A_; i += 128) {
        float sum = 0.0f;
        for (int j = 0; j < A_; ++j) {
            float dot = 0.0f;
            #pragma unroll
            for (int t = 0; t < WIN_; ++t) dot += xc[i][t] * xc[j][t];
            float c = dot * dinv[i] * dinv[j];
            sum += 1.0f - fabsf(c) + ((i == j) ? 1.0f : 0.0f);
        }
        rs[i] = rsqrtf(sum);
    }
    __syncthreads();

    float* out = nadj + (size_t)g * A_ * A_;
    for (int idx = tid; idx < A_ * A_; idx += 128) {
        int i = idx / A_, j = idx - i * A_;
        float dot = 0.0f;
        #pragma unroll
        for (int t = 0; t < WIN_; ++t) dot += xc[i][t] * xc[j][t];
        float c = dot * dinv[i] * dinv[j];
        float a = 1.0f - fabsf(c) + ((i == j) ? 1.0f : 0.0f);
        out[idx] = rs[i] * a * rs[j];
    }
}

// ---------------------------------------------------------------------------
// Kernel 2: per-group aggregation  t[g] = norm_adj[g](100x100) @ h[g](100xF)
// One wave32 per 32(M) x 32(N) tile: 2 A-frags x 2 B-frags -> 4 acc chains.
// Padded A rows (>= 100) are CLAMPED in-bounds (row m of A only influences
// row m of D; rows >= 100 are never stored) -> branch-free inner loop.
// Fragment layouts per CDNA5 ISA 7.12.2:
//   A (16x4 f32):  lane<16 -> M=lane, K={k,k+1}; lane>=16 -> M=lane-16, K={k+2,k+3}
//   B (4x16 f32):  mirrors A with N=lane&15
//   C/D (16x16):   col=lane&15, row = v + 8*(lane>>4)
// ---------------------------------------------------------------------------
template<int F>
__global__ __launch_bounds__(32) void agg_kernel(const float* __restrict__ nadj,
                                                 const float* __restrict__ hin,
                                                 float* __restrict__ tout) {
    const int g    = blockIdx.x;
    const int m0   = blockIdx.y * 32;
    const int n0   = blockIdx.z * 32;
    const int lane = threadIdx.x;
    const int lh   = lane >> 4;     // 0/1: which K half-pair
    const int ll   = lane & 15;

    const int arow0 = m0 + ll;
    const int arow1 = m0 + 16 + ll;
    const int rowc0 = (arow0 < A_) ? arow0 : (A_ - 1);   // in-bounds clamp
    const int rowc1 = (arow1 < A_) ? arow1 : (A_ - 1);
    const int bcol  = n0 + ll;

    const float* Ag = nadj + (size_t)g * A_ * A_;
    const float* ap0 = Ag + (size_t)rowc0 * A_ + lh * 2;
    const float* ap1 = Ag + (size_t)rowc1 * A_ + lh * 2;
    const float* bp  = hin + (size_t)g * A_ * F + (size_t)(lh * 2) * F + bcol;

    v8f acc00 = {}, acc01 = {}, acc10 = {}, acc11 = {};
    #pragma unroll 5
    for (int k = 0; k < A_; k += 4) {            // 25 iterations (K=100)
        const v2f a0 = *(const v2f*)(ap0 + k);   // 8B-aligned b64 loads
        const v2f a1 = *(const v2f*)(ap1 + k);
        v2f b0, b1;
        b0.x = bp[(size_t)k * F];
        b0.y = bp[(size_t)(k + 1) * F];
        b1.x = bp[(size_t)k * F + 16];
        b1.y = bp[(size_t)(k + 1) * F + 16];
        acc00 = __builtin_amdgcn_wmma_f32_16x16x4_f32(false, a0, false, b0,
                                                      (short)0, acc00, false, false);
        acc01 = __builtin_amdgcn_wmma_f32_16x16x4_f32(false, a0, false, b1,
                                                      (short)0, acc01, false, false);
        acc10 = __builtin_amdgcn_wmma_f32_16x16x4_f32(false, a1, false, b0,
                                                      (short)0, acc10, false, false);
        acc11 = __builtin_amdgcn_wmma_f32_16x16x4_f32(false, a1, false, b1,
                                                      (short)0, acc11, false, false);
    }

    float* Og = tout + (size_t)g * A_ * F;
    if (m0 + 32 <= A_) {
        // uniform fast path: one base pointer, compile-time immediate offsets
        float* base = Og + (size_t)(m0 + 8 * lh) * F + bcol;
        #pragma unroll
        for (int v = 0; v < 8; ++v) {
            base[(size_t)v * F]             = acc00[v];
            base[(size_t)v * F + 16]        = acc01[v];
            base[(size_t)(v + 16) * F]      = acc10[v];
            base[(size_t)(v + 16) * F + 16] = acc11[v];
        }
    } else {
        // last m-tile (m0 = 96): guarded stores
        #pragma unroll
        for (int v = 0; v < 8; ++v) {
            const int r0 = m0 + v + 8 * lh;
            const int r1 = r0 + 16;
            if (r0 < A_) {
                Og[(size_t)r0 * F + bcol]      = acc00[v];
                Og[(size_t)r0 * F + bcol + 16] = acc01[v];
            }
            if (r1 < A_) {
                Og[(size_t)r1 * F + bcol]      = acc10[v];
                Og[(size_t)r1 * F + bcol + 16] = acc11[v];
            }
        }
    }
}

// ---------------------------------------------------------------------------
// Kernel 3: batched dense  hout = relu(tin(RxK) @ W(Kx128) + b)
// R = 12800 (multiple of 32). One wave32 per 32x32 tile; K compile-time.
// ---------------------------------------------------------------------------
template<int K>
__global__ __launch_bounds__(32) void dense_kernel(const float* __restrict__ tin,
                                                   const float* __restrict__ W,
                                                   const float* __restrict__ bias,
                                                   float* __restrict__ hout) {
    const int m0   = blockIdx.x * 32;
    const int n0   = blockIdx.y * 32;
    const int lane = threadIdx.x;
    const int lh   = lane >> 4;
    const int ll   = lane & 15;

    const int bcol = n0 + ll;

    const float* ap0 = tin + (size_t)(m0 + ll) * K + lh * 2;
    const float* ap1 = ap0 + (size_t)16 * K;
    const float* wp  = W + (size_t)(lh * 2) * H_ + bcol;

    v8f acc00 = {}, acc01 = {}, acc10 = {}, acc11 = {};
    #pragma unroll 4
    for (int k = 0; k < K; k += 4) {
        if (k + 16 < K)   // prefetch ahead on the shared weight slab
            __builtin_prefetch(wp + (size_t)(k + 16) * H_, 0, 3);
        const v2f a0 = *(const v2f*)(ap0 + k);   // 8B-aligned b64 loads
        const v2f a1 = *(const v2f*)(ap1 + k);
        v2f b0, b1;
        b0.x = wp[(size_t)k * H_];
        b0.y = wp[(size_t)(k + 1) * H_];
        b1.x = wp[(size_t)k * H_ + 16];
        b1.y = wp[(size_t)(k + 1) * H_ + 16];
        acc00 = __builtin_amdgcn_wmma_f32_16x16x4_f32(false, a0, false, b0,
                                                      (short)0, acc00, false, false);
        acc01 = __builtin_amdgcn_wmma_f32_16x16x4_f32(false, a0, false, b1,
                                                      (short)0, acc01, false, false);
        acc10 = __builtin_amdgcn_wmma_f32_16x16x4_f32(false, a1, false, b0,
                                                      (short)0, acc10, false, false);
        acc11 = __builtin_amdgcn_wmma_f32_16x16x4_f32(false, a1, false, b1,
                                                      (short)0, acc11, false, false);
    }

    const float bb0 = bias[bcol];
    const float bb1 = bias[bcol + 16];
    float* base = hout + (size_t)(m0 + 8 * lh) * H_ + bcol;
    #pragma unroll
    for (int v = 0; v < 8; ++v) {
        base[(size_t)v * H_]             = fmaxf(acc00[v] + bb0, 0.0f);
        base[(size_t)v * H_ + 16]        = fmaxf(acc01[v] + bb1, 0.0f);
        base[(size_t)(v + 16) * H_]      = fmaxf(acc10[v] + bb0, 0.0f);
        base[(size_t)(v + 16) * H_ + 16] = fmaxf(acc11[v] + bb1, 0.0f);
    }
}

// ---------------------------------------------------------------------------
// Kernel 4: fused conv2d_1 (1x3, 1->128 ch, relu) + conv2d_2 (1x3, 128->1 ch)
// out[n,w] = cb2 + d0(w-1) + d1(w) + d2(w+1), where for position w':
//   s_c(w') = relu(h[n,w'-1]*cw1[c,0] + h[n,w']*cw1[c,1] + h[n,w'+1]*cw1[c,2] + cb1[c])
//   dk(w')  = sum_c s_c(w') * cw2[c,k]
// Thread tid == w'. 8 rows per block to amortize weight loads.
// ---------------------------------------------------------------------------
#define CONV_RPB 8

__global__ __launch_bounds__(128) void conv_kernel(const float* __restrict__ h,
                                                   const float* __restrict__ cw1,
                                                   const float* __restrict__ cb1,
                                                   const float* __restrict__ cw2,
                                                   const float* __restrict__ cb2,
                                                   float* __restrict__ out) {
    __shared__ float w1a[H_], w1b[H_], w1c[H_], b1s[H_];
    __shared__ float w2a[H_], w2b[H_], w2c[H_];
    __shared__ float hrow[H_ + 2];
    __shared__ float d0[H_], d1[H_], d2[H_];

    const int tid = threadIdx.x;
    w1a[tid] = cw1[tid * 3 + 0];
    w1b[tid] = cw1[tid * 3 + 1];
    w1c[tid] = cw1[tid * 3 + 2];
    b1s[tid] = cb1[tid];
    w2a[tid] = cw2[tid * 3 + 0];
    w2b[tid] = cw2[tid * 3 + 1];
    w2c[tid] = cw2[tid * 3 + 2];
    const float c2b = cb2[0];

    const int row0 = blockIdx.x * CONV_RPB;
    for (int r = 0; r < CONV_RPB; ++r) {
        const int n = row0 + r;
        __syncthreads();
        hrow[tid + 1] = h[(size_t)n * H_ + tid];
        if (tid == 0) { hrow[0] = 0.0f; hrow[H_ + 1] = 0.0f; }
        __syncthreads();

        const float hm = hrow[tid];
        const float h0 = hrow[tid + 1];
        const float hp = hrow[tid + 2];
        float s0 = 0.0f, s1 = 0.0f, s2 = 0.0f;
        #pragma unroll 4
        for (int c = 0; c < H_; ++c) {
            float s = fmaxf(fmaf(hm, w1a[c],
                          fmaf(h0, w1b[c],
                          fmaf(hp, w1c[c], b1s[c]))), 0.0f);
            s0 = fmaf(s, w2a[c], s0);
            s1 = fmaf(s, w2b[c], s1);
            s2 = fmaf(s, w2c[c], s2);
        }
        d0[tid] = s0; d1[tid] = s1; d2[tid] = s2;
        __syncthreads();

        float o = c2b + d1[tid];
        if (tid > 0)       o += d0[tid - 1];
        if (tid < H_ - 1)  o += d2[tid + 1];
        out[(size_t)n * H_ + tid] = o;
    }
}

// ---------------------------------------------------------------------------
// Launch
// ---------------------------------------------------------------------------
extern "C" void kernel_launch(void* const* d_in, const int* in_sizes, int n_in,
                              void* d_out, int out_size, void* d_ws, size_t ws_size,
                              hipStream_t stream) {
    const float* x   = (const float*)d_in[0];
    const float* W1  = (const float*)d_in[1];
    const float* b1  = (const float*)d_in[2];
    const float* W2  = (const float*)d_in[3];
    const float* b2  = (const float*)d_in[4];
    const float* W3  = (const float*)d_in[5];
    const float* b3  = (const float*)d_in[6];
    const float* cw1 = (const float*)d_in[7];
    const float* cb1 = (const float*)d_in[8];
    const float* cw2 = (const float*)d_in[9];
    const float* cb2 = (const float*)d_in[10];
    float* out = (float*)d_out;

    // Workspace layout (fp32):
    //   nadj : G*A*A          = 1,280,000
    //   tbuf : G*A*F1 (max)   = 2,048,000
    //   hA   : G*A*H          = 1,638,400
    //   hB   : G*A*H          = 1,638,400   -> ~26.4 MB total
    float* nadj = (float*)d_ws;
    float* tbuf = nadj + (size_t)G_ * A_ * A_;
    float* hA   = tbuf + (size_t)G_ * A_ * F1_;
    float* hB   = hA   + (size_t)N_ * H_;

    const int MT32 = (A_ + 31) / 32;              // 4 m-tiles per group
    const dim3 denseGrid(N_ / 32, H_ / 32);       // 400 x 4 waves

    corr_kernel<<<G_, 128, 0, stream>>>(x, nadj);

    // layer 1: feats = x reshaped [G,100,160]
    agg_kernel<F1_><<<dim3(G_, MT32, F1_ / 32), 32, 0, stream>>>(nadj, x, tbuf);
    dense_kernel<F1_><<<denseGrid, 32, 0, stream>>>(tbuf, W1, b1, hA);

    // layer 2
    agg_kernel<H_><<<dim3(G_, MT32, H_ / 32), 32, 0, stream>>>(nadj, hA, tbuf);
    dense_kernel<H_><<<denseGrid, 32, 0, stream>>>(tbuf, W2, b2, hB);

    // layer 3
    agg_kernel<H_><<<dim3(G_, MT32, H_ / 32), 32, 0, stream>>>(nadj, hB, tbuf);
    dense_kernel<H_><<<denseGrid, 32, 0, stream>>>(tbuf, W3, b3, hA);

    // fused conv pair -> [12800, 128]
    conv_kernel<<<N_ / CONV_RPB, 128, 0, stream>>>(hA, cw1, cb1, cw2, cb2, out);
}